// FixedKAN_48498770706594
// MI455X (gfx1250) — compile-verified
//
#include <hip/hip_runtime.h>

typedef __attribute__((ext_vector_type(2))) float v2f;
typedef __attribute__((ext_vector_type(4))) float v4f;
typedef __attribute__((ext_vector_type(8))) float v8f;

#define BM 128
#define BN 128
#define BK 16
#define SAP 9     // A row stride in float2 units (8 kpairs + 1 pad)
#define SBP 144   // B kpair-row stride in float2 (128 + 16 pad; 2*SBP mod 64 == 32
                  // -> the two half-wave fragment reads hit disjoint bank-pair sets)

static __device__ __forceinline__ v2f mkv2(float a, float b) {
  v2f r; r.x = a; r.y = b; return r;
}

#define WMMA_F32(A, B, C) \
  __builtin_amdgcn_wmma_f32_16x16x4_f32(false, (A), false, (B), (short)0, (C), false, false)

// ---------------------------------------------------------------------------
// Kernel 1: alpha = X[M,K] . W[N,K]^T + bn[N]  (NT GEMM, f32 WMMA)
//           Y = sum_d coef[n,d] * T_d(alpha)   (fused Chebyshev epilogue)
// ---------------------------------------------------------------------------
__global__ __launch_bounds__(256)
void kan_alpha_cheb(const float* __restrict__ X,
                    const float* __restrict__ W,
                    const float* __restrict__ bn,
                    const float* __restrict__ coef,   // [N, 8]
                    float* __restrict__ Y,
                    int M, int N, int K)
{
  __shared__ v2f sA[2][BM * SAP];          // [m][kpair] pairs (k, k+1)
  __shared__ v2f sB[2][(BK / 2) * SBP];    // [kpair][n] pairs (k, k+1)

  const int tid  = threadIdx.x;
  const int wave = tid >> 5;
  const int lane = tid & 31;
  const int half = lane >> 4;
  const int l16  = lane & 15;
  const int wm   = wave & 3;               // 4 waves along M -> 32 rows each
  const int wn   = wave >> 2;              // 2 waves along N -> 64 cols each

  const int m0 = blockIdx.y * BM;
  const int n0 = blockIdx.x * BN;

  v8f acc[2][4] = {};
  v4f ar[2], br[2];

  // Both A and B (NT) tiles are [128 rows][16 k], K contiguous -> same staging map.
  auto gload = [&](int k0) {
    #pragma unroll
    for (int i = 0; i < 2; ++i) {
      const int f  = tid + i * 256;
      const int k4 = f & 3;
      const int r  = f >> 2;
      ar[i] = *(const v4f*)(X + (size_t)(m0 + r) * K + k0 + k4 * 4);
      br[i] = *(const v4f*)(W + (size_t)(n0 + r) * K + k0 + k4 * 4);
    }
  };
  auto lstore = [&](int buf) {
    #pragma unroll
    for (int i = 0; i < 2; ++i) {
      const int f  = tid + i * 256;
      const int k4 = f & 3;
      const int r  = f >> 2;
      sA[buf][r * SAP + 2 * k4    ] = mkv2(ar[i].x, ar[i].y);
      sA[buf][r * SAP + 2 * k4 + 1] = mkv2(ar[i].z, ar[i].w);
      sB[buf][(2 * k4    ) * SBP + r] = mkv2(br[i].x, br[i].y);
      sB[buf][(2 * k4 + 1) * SBP + r] = mkv2(br[i].z, br[i].w);
    }
  };
  auto compute = [&](int buf) {
    const v2f* A0 = &sA[buf][(wm * 32 + l16) * SAP];
    const v2f* Bb = &sB[buf][wn * 64 + l16];
    #pragma unroll
    for (int kk = 0; kk < 8; kk += 2) {    // kpair steps
      const int kp = kk + half;
      v2f a0 = A0[kp];
      v2f a1 = A0[16 * SAP + kp];
      const v2f* Br = Bb + kp * SBP;
      v2f b0 = Br[0], b1 = Br[16], b2 = Br[32], b3 = Br[48];
      acc[0][0] = WMMA_F32(a0, b0, acc[0][0]);
      acc[0][1] = WMMA_F32(a0, b1, acc[0][1]);
      acc[0][2] = WMMA_F32(a0, b2, acc[0][2]);
      acc[0][3] = WMMA_F32(a0, b3, acc[0][3]);
      acc[1][0] = WMMA_F32(a1, b0, acc[1][0]);
      acc[1][1] = WMMA_F32(a1, b1, acc[1][1]);
      acc[1][2] = WMMA_F32(a1, b2, acc[1][2]);
      acc[1][3] = WMMA_F32(a1, b3, acc[1][3]);
    }
  };

  const int nChunks = K / BK;
  gload(0);
  lstore(0);
  __syncthreads();
  for (int c = 0; c < nChunks; ++c) {
    const int cur = c & 1;
    if (c + 1 < nChunks) gload((c + 1) * BK);
    compute(cur);
    if (c + 1 < nChunks) lstore(cur ^ 1);
    __syncthreads();
  }

  // Fused epilogue: bias + Chebyshev T0..T7 weighted by coef
  #pragma unroll
  for (int mi = 0; mi < 2; ++mi) {
    const int mBase = m0 + wm * 32 + mi * 16 + 8 * half;
    #pragma unroll
    for (int nt = 0; nt < 4; ++nt) {
      const int gn = n0 + wn * 64 + nt * 16 + l16;
      const float bias = bn[gn];
      v4f c0 = *(const v4f*)(coef + (size_t)gn * 8);
      v4f c1 = *(const v4f*)(coef + (size_t)gn * 8 + 4);
      const float c[8] = {c0.x, c0.y, c0.z, c0.w, c1.x, c1.y, c1.z, c1.w};
      #pragma unroll
      for (int r = 0; r < 8; ++r) {
        float al  = acc[mi][nt][r] + bias;
        float tm1 = 1.0f, t = al;
        float y = fmaf(c[1], al, c[0]);
        #pragma unroll
        for (int d = 2; d < 8; ++d) {
          float tn = fmaf(2.0f * al, t, -tm1);
          y = fmaf(c[d], tn, y);
          tm1 = t; t = tn;
        }
        Y[(size_t)(mBase + r) * N + gn] = y;
      }
    }
  }
}

// ---------------------------------------------------------------------------
// Kernel 2: H = Yin[M,K] . CW[K,N] + cb[N]  (NN GEMM, f32 WMMA)
// ---------------------------------------------------------------------------
__global__ __launch_bounds__(256)
void kan_combine(const float* __restrict__ Yin,
                 const float* __restrict__ CW,    // [K, N]
                 const float* __restrict__ cb,
                 float* __restrict__ Hout,
                 int M, int N, int K)
{
  __shared__ v2f sA[2][BM * SAP];
  __shared__ v2f sB[2][(BK / 2) * SBP];

  const int tid  = threadIdx.x;
  const int wave = tid >> 5;
  const int lane = tid & 31;
  const int half = lane >> 4;
  const int l16  = lane & 15;
  const int wm   = wave & 3;
  const int wn   = wave >> 2;

  const int m0 = blockIdx.y * BM;
  const int n0 = blockIdx.x * BN;

  v8f acc[2][4] = {};
  v4f ar[2], ba, bb;

  auto gload = [&](int k0) {
    #pragma unroll
    for (int i = 0; i < 2; ++i) {
      const int f  = tid + i * 256;
      const int k4 = f & 3;
      const int r  = f >> 2;
      ar[i] = *(const v4f*)(Yin + (size_t)(m0 + r) * K + k0 + k4 * 4);
    }
    // B (NN): read rows k=2kp and 2kp+1 so K-pairs can be packed on store.
    const int kp = tid >> 5;
    const int n4 = tid & 31;
    ba = *(const v4f*)(CW + (size_t)(k0 + 2 * kp    ) * N + n0 + 4 * n4);
    bb = *(const v4f*)(CW + (size_t)(k0 + 2 * kp + 1) * N + n0 + 4 * n4);
  };
  auto lstore = [&](int buf) {
    #pragma unroll
    for (int i = 0; i < 2; ++i) {
      const int f  = tid + i * 256;
      const int k4 = f & 3;
      const int r  = f >> 2;
      sA[buf][r * SAP + 2 * k4    ] = mkv2(ar[i].x, ar[i].y);
      sA[buf][r * SAP + 2 * k4 + 1] = mkv2(ar[i].z, ar[i].w);
    }
    const int kp = tid >> 5;
    const int n4 = tid & 31;
    v2f* d = &sB[buf][kp * SBP + 4 * n4];
    d[0] = mkv2(ba.x, bb.x);
    d[1] = mkv2(ba.y, bb.y);
    d[2] = mkv2(ba.z, bb.z);
    d[3] = mkv2(ba.w, bb.w);
  };
  auto compute = [&](int buf) {
    const v2f* A0 = &sA[buf][(wm * 32 + l16) * SAP];
    const v2f* Bb = &sB[buf][wn * 64 + l16];
    #pragma unroll
    for (int kk = 0; kk < 8; kk += 2) {
      const int kp = kk + half;
      v2f a0 = A0[kp];
      v2f a1 = A0[16 * SAP + kp];
      const v2f* Br = Bb + kp * SBP;
      v2f b0 = Br[0], b1 = Br[16], b2 = Br[32], b3 = Br[48];
      acc[0][0] = WMMA_F32(a0, b0, acc[0][0]);
      acc[0][1] = WMMA_F32(a0, b1, acc[0][1]);
      acc[0][2] = WMMA_F32(a0, b2, acc[0][2]);
      acc[0][3] = WMMA_F32(a0, b3, acc[0][3]);
      acc[1][0] = WMMA_F32(a1, b0, acc[1][0]);
      acc[1][1] = WMMA_F32(a1, b1, acc[1][1]);
      acc[1][2] = WMMA_F32(a1, b2, acc[1][2]);
      acc[1][3] = WMMA_F32(a1, b3, acc[1][3]);
    }
  };

  const int nChunks = K / BK;
  gload(0);
  lstore(0);
  __syncthreads();
  for (int c = 0; c < nChunks; ++c) {
    const int cur = c & 1;
    if (c + 1 < nChunks) gload((c + 1) * BK);
    compute(cur);
    if (c + 1 < nChunks) lstore(cur ^ 1);
    __syncthreads();
  }

  #pragma unroll
  for (int mi = 0; mi < 2; ++mi) {
    const int mBase = m0 + wm * 32 + mi * 16 + 8 * half;
    #pragma unroll
    for (int nt = 0; nt < 4; ++nt) {
      const int gn = n0 + wn * 64 + nt * 16 + l16;
      const float bias = cb[gn];
      #pragma unroll
      for (int r = 0; r < 8; ++r) {
        Hout[(size_t)(mBase + r) * N + gn] = acc[mi][nt][r] + bias;
      }
    }
  }
}

// ---------------------------------------------------------------------------
// Host launch: 3 layers x (alpha+cheb, combine); intermediates ping-pong in ws
// (2 x 16384*512 f32 = 64 MB; both live in the 192 MB L2).
// ---------------------------------------------------------------------------
extern "C" void kernel_launch(void* const* d_in, const int* in_sizes, int n_in,
                              void* d_out, int out_size, void* d_ws, size_t ws_size,
                              hipStream_t stream) {
  (void)in_sizes; (void)n_in; (void)out_size; (void)ws_size;

  const float* x     = (const float*)d_in[0];
  const float* wn0   = (const float*)d_in[1];
  const float* bn0   = (const float*)d_in[2];
  const float* coef0 = (const float*)d_in[3];
  const float* cw0   = (const float*)d_in[4];
  const float* cb0   = (const float*)d_in[5];
  const float* wn1   = (const float*)d_in[6];
  const float* bn1   = (const float*)d_in[7];
  const float* coef1 = (const float*)d_in[8];
  const float* cw1   = (const float*)d_in[9];
  const float* cb1   = (const float*)d_in[10];
  const float* wn2   = (const float*)d_in[11];
  const float* bn2   = (const float*)d_in[12];
  const float* coef2 = (const float*)d_in[13];
  const float* cw2   = (const float*)d_in[14];
  const float* cb2   = (const float*)d_in[15];

  const int B = 16384;
  float* ws0 = (float*)d_ws;             // [B, 512]
  float* ws1 = ws0 + (size_t)B * 512;    // [B, 512]
  float* out = (float*)d_out;            // [B, 128]

  const dim3 blk(256);
  const dim3 g512(512 / BN, B / BM);     // (4, 128)
  const dim3 g128(128 / BN, B / BM);     // (1, 128)

  // Layer 0: 256 -> 512
  kan_alpha_cheb<<<g512, blk, 0, stream>>>(x,   wn0, bn0, coef0, ws0, B, 512, 256);
  kan_combine  <<<g512, blk, 0, stream>>>(ws0, cw0, cb0, ws1, B, 512, 512);
  // Layer 1: 512 -> 512
  kan_alpha_cheb<<<g512, blk, 0, stream>>>(ws1, wn1, bn1, coef1, ws0, B, 512, 512);
  kan_combine  <<<g512, blk, 0, stream>>>(ws0, cw1, cb1, ws1, B, 512, 512);
  // Layer 2: 512 -> 128
  kan_alpha_cheb<<<g128, blk, 0, stream>>>(ws1, wn2, bn2, coef2, ws0, B, 128, 512);
  kan_combine  <<<g128, blk, 0, stream>>>(ws0, cw2, cb2, out, B, 128, 128);
}